// GraphSAGE_44813688767214
// MI455X (gfx1250) — compile-verified
//
#include <hip/hip_runtime.h>
#include <hip/hip_bf16.h>
#include <math.h>

// ---------------------------------------------------------------------------
// GraphSAGE forward on MI455X (gfx1250, wave32, WMMA).
//  - node GEMMs via V_WMMA_F32_16X16X4_F32 (exact f32, matches reference math)
//  - weights repacked once per launch into fragment-native layout so each
//    B fragment is a single coalesced global_load_b64
//  - each wave owns a 32x16 output tile (2 accumulators) -> every B fragment
//    feeds 2 WMMAs
//  - edge aggregation via global f32 atomics (x/h1 are L2-resident: 51/102 MB
//    vs 192 MB L2, so gathers hit cache)
//  - h2 never materialized: layer-2 epilogue pools directly with atomics
// ---------------------------------------------------------------------------

#define N_NODES   100000
#define N_EDGES   800000
#define IN_C      128
#define HID       256
#define N_GRAPHS  64
#define BN_EPS    1e-5f

typedef __attribute__((ext_vector_type(2))) float v2f;
typedef __attribute__((ext_vector_type(8))) float v8f;

// ---------------------------------------------------------------------------
// Repack W[K][HID] -> Wp[q][h][col][v], q=k/4, h=(k/2)&1, v=k&1.
// Lane (half h, col) then loads its v2f B fragment for k-quad q as one b64:
//   Wp + (((q*2 + h)*HID + col) * 2)
// ---------------------------------------------------------------------------
__global__ __launch_bounds__(256) void repack_w(
    const float* __restrict__ W, float* __restrict__ Wp, int K) {
  int i = blockIdx.x * blockDim.x + threadIdx.x;
  if (i >= K * HID) return;
  int k = i >> 8;          // HID == 256
  int c = i & 255;
  int q = k >> 2;
  int h = (k >> 1) & 1;
  int v = k & 1;
  Wp[((((q << 1) + h) << 8) + c) * 2 + v] = W[i];
}

// ---------------------------------------------------------------------------
// Edge scatter, layer 1: summed[dst] += x[src]; deg[dst] += 1
// ---------------------------------------------------------------------------
__global__ __launch_bounds__(256) void scatter_l1(
    const float* __restrict__ x, const int* __restrict__ ei,
    float* __restrict__ summed, float* __restrict__ deg) {
  long long idx = (long long)blockIdx.x * blockDim.x + threadIdx.x;
  if (idx >= (long long)N_EDGES * (IN_C / 4)) return;
  int e = (int)(idx >> 5);
  int f = ((int)idx & 31) << 2;
  int s = ei[e];
  int d = ei[N_EDGES + e];
  const float4 v = *(const float4*)(x + (size_t)s * IN_C + f);
  float* o = summed + (size_t)d * IN_C + f;
  atomicAdd(o + 0, v.x); atomicAdd(o + 1, v.y);
  atomicAdd(o + 2, v.z); atomicAdd(o + 3, v.w);
  if ((idx & 31) == 0) atomicAdd(deg + d, 1.0f);
}

// ---------------------------------------------------------------------------
// Edge scatter, layer 2: summed[dst] += h1[src]
// ---------------------------------------------------------------------------
__global__ __launch_bounds__(256) void scatter_l2(
    const float* __restrict__ h1, const int* __restrict__ ei,
    float* __restrict__ summed) {
  long long idx = (long long)blockIdx.x * blockDim.x + threadIdx.x;
  if (idx >= (long long)N_EDGES * (HID / 4)) return;
  int e = (int)(idx >> 6);
  int f = ((int)idx & 63) << 2;
  int s = ei[e];
  int d = ei[N_EDGES + e];
  const float4 v = *(const float4*)(h1 + (size_t)s * HID + f);
  float* o = summed + (size_t)d * HID + f;
  atomicAdd(o + 0, v.x); atomicAdd(o + 1, v.y);
  atomicAdd(o + 2, v.z); atomicAdd(o + 3, v.w);
}

// ---------------------------------------------------------------------------
__global__ __launch_bounds__(256) void count_kernel(
    const int* __restrict__ batch, float* __restrict__ cnt) {
  int i = blockIdx.x * blockDim.x + threadIdx.x;
  if (i < N_NODES) atomicAdd(&cnt[batch[i]], 1.0f);
}

// ---------------------------------------------------------------------------
// Layer 1: h1 = ReLU(BN( (summed/deg) @ W1a + x @ W1r + b1 ))
// Block = 512 threads = 16 waves; block owns 32 rows x 256 cols; each wave a
// 32x16 tile (2 accumulators) -> each B fragment (b64) feeds 2 WMMAs.
// ---------------------------------------------------------------------------
#define PITCH1 132   // 128 + 4; pitch % 64 == 4 -> conflict-free column reads
__global__ __launch_bounds__(512) void sage1_gemm(
    const float* __restrict__ x, const float* __restrict__ summed,
    const float* __restrict__ deg,
    const float* __restrict__ W1a_p, const float* __restrict__ W1r_p,
    const float* __restrict__ b1,
    const float* __restrict__ gamma, const float* __restrict__ beta,
    const float* __restrict__ mean, const float* __restrict__ var,
    float* __restrict__ h1) {
  __shared__ float sAgg[32 * PITCH1];
  __shared__ float sX[32 * PITCH1];
  const int tid  = threadIdx.x;
  const int row0 = blockIdx.x * 32;          // N_NODES == 3125*32 exactly

  for (int i = tid; i < 32 * (IN_C / 4); i += 512) {
    int r = i >> 5;                          // 32 float4 per row
    int c = (i & 31) << 2;
    float rdeg = 1.0f / fmaxf(deg[row0 + r], 1.0f);
    float4 sv = *(const float4*)(summed + (size_t)(row0 + r) * IN_C + c);
    float4 xv = *(const float4*)(x      + (size_t)(row0 + r) * IN_C + c);
    float* pa = sAgg + r * PITCH1 + c;
    pa[0] = sv.x * rdeg; pa[1] = sv.y * rdeg; pa[2] = sv.z * rdeg; pa[3] = sv.w * rdeg;
    float* px = sX + r * PITCH1 + c;
    px[0] = xv.x; px[1] = xv.y; px[2] = xv.z; px[3] = xv.w;
  }
  __syncthreads();

  const int wave = tid >> 5;
  const int lane = tid & 31;
  const int half = lane >> 4;
  const int mrow = lane & 15;
  const int col  = wave * 16 + mrow;         // output column 0..255

  v8f acc0 = {}, acc1 = {};
  for (int q = 0; q < IN_C / 4; ++q) {
    int ka = q * 4 + 2 * half;
    v2f a0g = *(const v2f*)(sAgg + mrow * PITCH1 + ka);
    v2f a1g = *(const v2f*)(sAgg + (mrow + 16) * PITCH1 + ka);
    v2f a0x = *(const v2f*)(sX + mrow * PITCH1 + ka);
    v2f a1x = *(const v2f*)(sX + (mrow + 16) * PITCH1 + ka);
    size_t bo = ((size_t)((q << 1) + half) * HID + col) * 2;
    v2f bA = *(const v2f*)(W1a_p + bo);
    v2f bR = *(const v2f*)(W1r_p + bo);
    acc0 = __builtin_amdgcn_wmma_f32_16x16x4_f32(false, a0g, false, bA,
                                                 (short)0, acc0, false, false);
    acc1 = __builtin_amdgcn_wmma_f32_16x16x4_f32(false, a1g, false, bA,
                                                 (short)0, acc1, false, false);
    acc0 = __builtin_amdgcn_wmma_f32_16x16x4_f32(false, a0x, false, bR,
                                                 (short)0, acc0, false, false);
    acc1 = __builtin_amdgcn_wmma_f32_16x16x4_f32(false, a1x, false, bR,
                                                 (short)0, acc1, false, false);
  }

  // fused epilogue: +bias, BatchNorm (eval), ReLU
  float scale = gamma[col] * rsqrtf(var[col] + BN_EPS);
  float shift = beta[col] - mean[col] * scale;
  float bias  = b1[col];
#pragma unroll
  for (int v = 0; v < 8; ++v) {
    int m = v + 8 * half;
    float v0 = (acc0[v] + bias) * scale + shift;
    float v1 = (acc1[v] + bias) * scale + shift;
    h1[(size_t)(row0 + m) * HID + col]      = fmaxf(v0, 0.0f);
    h1[(size_t)(row0 + 16 + m) * HID + col] = fmaxf(v1, 0.0f);
  }
}

// ---------------------------------------------------------------------------
// Layer 2: h2 = (summed/deg) @ W2a + h1 @ W2r + b2, pooled in epilogue:
// pooled[batch[row]] += h2[row]  (h2 never written to memory).
// ---------------------------------------------------------------------------
#define PITCH2 260   // 256 + 4
__global__ __launch_bounds__(512) void sage2_gemm(
    const float* __restrict__ h1, const float* __restrict__ summed,
    const float* __restrict__ deg,
    const float* __restrict__ W2a_p, const float* __restrict__ W2r_p,
    const float* __restrict__ b2, const int* __restrict__ batch,
    float* __restrict__ pooled) {
  __shared__ float sAgg[32 * PITCH2];
  __shared__ float sX[32 * PITCH2];
  const int tid  = threadIdx.x;
  const int row0 = blockIdx.x * 32;

  for (int i = tid; i < 32 * (HID / 4); i += 512) {
    int r = i >> 6;                          // 64 float4 per row
    int c = (i & 63) << 2;
    float rdeg = 1.0f / fmaxf(deg[row0 + r], 1.0f);
    float4 sv = *(const float4*)(summed + (size_t)(row0 + r) * HID + c);
    float4 xv = *(const float4*)(h1     + (size_t)(row0 + r) * HID + c);
    float* pa = sAgg + r * PITCH2 + c;
    pa[0] = sv.x * rdeg; pa[1] = sv.y * rdeg; pa[2] = sv.z * rdeg; pa[3] = sv.w * rdeg;
    float* px = sX + r * PITCH2 + c;
    px[0] = xv.x; px[1] = xv.y; px[2] = xv.z; px[3] = xv.w;
  }
  __syncthreads();

  const int wave = tid >> 5;
  const int lane = tid & 31;
  const int half = lane >> 4;
  const int mrow = lane & 15;
  const int col  = wave * 16 + mrow;

  v8f acc0 = {}, acc1 = {};
  for (int q = 0; q < HID / 4; ++q) {
    int ka = q * 4 + 2 * half;
    v2f a0g = *(const v2f*)(sAgg + mrow * PITCH2 + ka);
    v2f a1g = *(const v2f*)(sAgg + (mrow + 16) * PITCH2 + ka);
    v2f a0x = *(const v2f*)(sX + mrow * PITCH2 + ka);
    v2f a1x = *(const v2f*)(sX + (mrow + 16) * PITCH2 + ka);
    size_t bo = ((size_t)((q << 1) + half) * HID + col) * 2;
    v2f bA = *(const v2f*)(W2a_p + bo);
    v2f bR = *(const v2f*)(W2r_p + bo);
    acc0 = __builtin_amdgcn_wmma_f32_16x16x4_f32(false, a0g, false, bA,
                                                 (short)0, acc0, false, false);
    acc1 = __builtin_amdgcn_wmma_f32_16x16x4_f32(false, a1g, false, bA,
                                                 (short)0, acc1, false, false);
    acc0 = __builtin_amdgcn_wmma_f32_16x16x4_f32(false, a0x, false, bR,
                                                 (short)0, acc0, false, false);
    acc1 = __builtin_amdgcn_wmma_f32_16x16x4_f32(false, a1x, false, bR,
                                                 (short)0, acc1, false, false);
  }

  float bias = b2[col];
#pragma unroll
  for (int v = 0; v < 8; ++v) {
    int m = v + 8 * half;
    int r0 = row0 + m;
    int r1 = row0 + 16 + m;
    atomicAdd(&pooled[(size_t)batch[r0] * HID + col], acc0[v] + bias);
    atomicAdd(&pooled[(size_t)batch[r1] * HID + col], acc1[v] + bias);
  }
}

// ---------------------------------------------------------------------------
// Head: g = pooled/cnt; g1 = relu(g@Wc1+bc1); logits = g1@Wc2+bc2; log_softmax
// ---------------------------------------------------------------------------
__global__ __launch_bounds__(256) void head_kernel(
    const float* __restrict__ pooled, const float* __restrict__ cnt,
    const float* __restrict__ Wc1, const float* __restrict__ bc1,
    const float* __restrict__ Wc2, const float* __restrict__ bc2,
    float* __restrict__ out) {
  __shared__ float g1[N_GRAPHS * 128];
  __shared__ float rc[N_GRAPHS];
  __shared__ float lg[N_GRAPHS * 2];
  int tid = threadIdx.x;
  if (tid < N_GRAPHS) rc[tid] = 1.0f / fmaxf(cnt[tid], 1.0f);
  __syncthreads();

  for (int o = tid; o < N_GRAPHS * 128; o += 256) {
    int i = o >> 7;
    int c = o & 127;
    float acc = 0.0f;
    for (int k = 0; k < HID; ++k)
      acc += pooled[(size_t)i * HID + k] * Wc1[(size_t)k * 128 + c];
    acc = acc * rc[i] + bc1[c];
    g1[o] = fmaxf(acc, 0.0f);
  }
  __syncthreads();

  if (tid < N_GRAPHS * 2) {
    int i = tid >> 1;
    int j = tid & 1;
    float acc = bc2[j];
    for (int k = 0; k < 128; ++k) acc += g1[i * 128 + k] * Wc2[k * 2 + j];
    lg[tid] = acc;
  }
  __syncthreads();

  if (tid < N_GRAPHS) {
    float l0 = lg[2 * tid], l1 = lg[2 * tid + 1];
    float m  = fmaxf(l0, l1);
    float lse = m + logf(expf(l0 - m) + expf(l1 - m));
    out[2 * tid]     = l0 - lse;
    out[2 * tid + 1] = l1 - lse;
  }
}

// ---------------------------------------------------------------------------
extern "C" void kernel_launch(void* const* d_in, const int* in_sizes, int n_in,
                              void* d_out, int out_size, void* d_ws, size_t ws_size,
                              hipStream_t stream) {
  const float* x     = (const float*)d_in[0];
  const int*   ei    = (const int*)  d_in[1];
  const int*   batch = (const int*)  d_in[2];
  const float* W1a   = (const float*)d_in[3];
  const float* W1r   = (const float*)d_in[4];
  const float* b1    = (const float*)d_in[5];
  const float* gamma = (const float*)d_in[6];
  const float* beta  = (const float*)d_in[7];
  const float* mean  = (const float*)d_in[8];
  const float* var   = (const float*)d_in[9];
  const float* W2a   = (const float*)d_in[10];
  const float* W2r   = (const float*)d_in[11];
  const float* b2    = (const float*)d_in[12];
  const float* Wc1   = (const float*)d_in[13];
  const float* bc1   = (const float*)d_in[14];
  const float* Wc2   = (const float*)d_in[15];
  const float* bc2   = (const float*)d_in[16];
  float* out = (float*)d_out;

  // workspace layout (floats)
  const size_t N_PAD = 100096;  // N_NODES rounded to 128
  float* ws     = (float*)d_ws;
  float* deg    = ws;                                   // N_NODES
  float* summed = ws + N_PAD;                           // N_NODES*HID (reused)
  float* h1     = summed + (size_t)N_NODES * HID;       // N_NODES*HID
  float* pooled = h1 + (size_t)N_NODES * HID;           // 64*HID
  float* cnt    = pooled + (size_t)N_GRAPHS * HID;      // 64
  float* w1a_p  = cnt + N_GRAPHS;                       // 128*256
  float* w1r_p  = w1a_p + IN_C * HID;                   // 128*256
  float* w2a_p  = w1r_p + IN_C * HID;                   // 256*256
  float* w2r_p  = w2a_p + HID * HID;                    // 256*256

  // ---- weight repack (fragment-native, coalesced b64 loads in GEMMs) ----
  repack_w<<<(IN_C * HID) / 256, 256, 0, stream>>>(W1a, w1a_p, IN_C);
  repack_w<<<(IN_C * HID) / 256, 256, 0, stream>>>(W1r, w1r_p, IN_C);
  repack_w<<<(HID * HID) / 256, 256, 0, stream>>>(W2a, w2a_p, HID);
  repack_w<<<(HID * HID) / 256, 256, 0, stream>>>(W2r, w2r_p, HID);

  // ---- layer 1 ----
  hipMemsetAsync(deg, 0, sizeof(float) * N_NODES, stream);
  hipMemsetAsync(summed, 0, sizeof(float) * (size_t)N_NODES * IN_C, stream);
  {
    long long t = (long long)N_EDGES * (IN_C / 4);
    scatter_l1<<<(unsigned)((t + 255) / 256), 256, 0, stream>>>(x, ei, summed, deg);
  }
  sage1_gemm<<<N_NODES / 32, 512, 0, stream>>>(x, summed, deg, w1a_p, w1r_p, b1,
                                               gamma, beta, mean, var, h1);

  // ---- layer 2 (+ fused pooling) ----
  hipMemsetAsync(summed, 0, sizeof(float) * (size_t)N_NODES * HID, stream);
  hipMemsetAsync(pooled, 0, sizeof(float) * ((size_t)N_GRAPHS * HID + N_GRAPHS), stream);
  {
    long long t = (long long)N_EDGES * (HID / 4);
    scatter_l2<<<(unsigned)((t + 255) / 256), 256, 0, stream>>>(h1, ei, summed);
  }
  count_kernel<<<(N_NODES + 255) / 256, 256, 0, stream>>>(batch, cnt);
  sage2_gemm<<<N_NODES / 32, 512, 0, stream>>>(h1, summed, deg, w2a_p, w2r_p, b2,
                                               batch, pooled);

  // ---- head ----
  head_kernel<<<1, 256, 0, stream>>>(pooled, cnt, Wc1, bc1, Wc2, bc2, out);
}